// Router_38757784879528
// MI455X (gfx1250) — compile-verified
//
#include <hip/hip_runtime.h>
#include <hip/hip_bf16.h>
#include <cstdint>

typedef __attribute__((ext_vector_type(2))) float v2f;
typedef __attribute__((ext_vector_type(8))) float v8f;
typedef __attribute__((ext_vector_type(4))) int   v4i;

// Pointer types matching the async-builtin prototype revealed by the compiler:
//   param0: v4i in addrspace(1) (global), param1: v4i in addrspace(3) (LDS)
typedef __attribute__((address_space(1))) v4i* gv4i_p;
typedef __attribute__((address_space(3))) v4i* lv4i_p;

#define TOKENS 32768
#define HIDDEN 4096
#define NEXP   128
#define TOPK   8
#define BM     128   // tokens per workgroup
#define KC     64    // K-chunk staged in LDS
#define LDA    68    // padded stride: 68*4B=272 keeps 16B alignment for B128; 4*(l%16)+2h covers 64 banks conflict-free
#define LLOG   130   // padded logits stride (even -> aligned v2f reads)
#define STILE  (BM * LDA)   // floats per staging tile (8704)

#if defined(__AMDGCN__) && __has_builtin(__builtin_amdgcn_global_load_async_to_lds_b128)
#define USE_ASYNC 1
#else
#define USE_ASYNC 0
#endif

__device__ __forceinline__ void wait_async0() {
#if __has_builtin(__builtin_amdgcn_s_wait_asynccnt)
    __builtin_amdgcn_s_wait_asynccnt(0);
#elif defined(__AMDGCN__)
    asm volatile("s_wait_asynccnt 0x0" ::: "memory");
#endif
}

// Stage one 128xKC fp32 tile pair (x rows + W rows) for K-offset k0 into LDS buffers.
__device__ __forceinline__ void stage_tiles(const float* __restrict__ x,
                                            const float* __restrict__ w,
                                            float* ldsA, float* ldsB,
                                            int tok0, int k0, int tid) {
#pragma unroll
    for (int i = 0; i < 8; ++i) {
        const int idx = tid + i * 256;      // 0..2047
        const int row = idx >> 4;           // 0..127
        const int c4  = (idx & 15) << 2;    // 0,4,...,60
        const float* ga = x + (size_t)(tok0 + row) * HIDDEN + k0 + c4;
        const float* gb = w + (size_t)row * HIDDEN + k0 + c4;
        float* la = &ldsA[row * LDA + c4];
        float* lb = &ldsB[row * LDA + c4];
#if USE_ASYNC
        __builtin_amdgcn_global_load_async_to_lds_b128(
            (gv4i_p)(uintptr_t)ga,
            (lv4i_p)(uint32_t)(uintptr_t)la,
            0, 0);
        __builtin_amdgcn_global_load_async_to_lds_b128(
            (gv4i_p)(uintptr_t)gb,
            (lv4i_p)(uint32_t)(uintptr_t)lb,
            0, 0);
#else
        const float4 va = *(const float4*)ga;
        const float4 vb = *(const float4*)gb;
        *(float4*)la = va;
        *(float4*)lb = vb;
#endif
    }
}

__global__ __launch_bounds__(256) void router_kernel(
    const float* __restrict__ x, const float* __restrict__ w,
    float* __restrict__ out_w, int* __restrict__ out_i)
{
    // Double-buffered staging: A0 B0 A1 B1, each 128x68 f32. Total 136 KB (<320 KB WGP LDS).
    // Logits (128x130 f32 = 65 KB) reuse the A0/B0 region after the K loop.
    __shared__ float lds[4 * STILE];

    const int tid  = threadIdx.x;
    const int wave = tid >> 5;
    const int lane = tid & 31;
    const int half = lane >> 4;
    const int l16  = lane & 15;
    const int tok0 = blockIdx.x * BM;

    v8f acc[8];
    const v8f vzero = {0.f, 0.f, 0.f, 0.f, 0.f, 0.f, 0.f, 0.f};
#pragma unroll
    for (int t = 0; t < 8; ++t) acc[t] = vzero;

    // Prologue: prefetch chunk 0 into buffer 0.
    stage_tiles(x, w, &lds[0], &lds[STILE], tok0, 0, tid);

    for (int k0 = 0, it = 0; k0 < HIDDEN; k0 += KC, ++it) {
        const int c = it & 1;
        float* curA = &lds[(2 * c + 0) * STILE];
        float* curB = &lds[(2 * c + 1) * STILE];

        wait_async0();       // own chunk-`it` loads complete
        __syncthreads();     // all waves' chunk-`it` loads complete; buf 1-c free for reuse

        // Prefetch next chunk into the other buffer; overlaps with compute below.
        if (k0 + KC < HIDDEN) {
            stage_tiles(x, w, &lds[(2 * (1 - c) + 0) * STILE], &lds[(2 * (1 - c) + 1) * STILE],
                        tok0, k0 + KC, tid);
        }

        // Wave `wave` owns tokens [wave*16, wave*16+16) x all 128 experts (8 N-tiles).
        const int abase = (wave * 16 + l16) * LDA + 2 * half;
#pragma unroll 2
        for (int kk = 0; kk < KC; kk += 4) {
            const v2f a = *(const v2f*)&curA[abase + kk];  // A[m][k0+kk+2h+{0,1}]
#pragma unroll
            for (int t = 0; t < 8; ++t) {
                const v2f b = *(const v2f*)&curB[(t * 16 + l16) * LDA + 2 * half + kk];
                acc[t] = __builtin_amdgcn_wmma_f32_16x16x4_f32(
                    false, a, false, b, (short)0, acc[t], false, false);
            }
        }
    }

    // Scatter logits to LDS (buffers 0-1 region; last compute chunk used buffer 1 -> disjoint).
    // C tile VGPR r -> M = r + 8*half, N = lane%16.
#pragma unroll
    for (int t = 0; t < 8; ++t) {
#pragma unroll
        for (int r = 0; r < 8; ++r) {
            lds[(wave * 16 + r + 8 * half) * LLOG + t * 16 + l16] = acc[t][r];
        }
    }
    __syncthreads();

    // One thread per token: softmax stats + stable top-8.
    if (tid < BM) {
        const float* rowp = &lds[tid * LLOG];
        float mx = -3.4e38f;
        for (int e = 0; e < NEXP; e += 2) {
            const v2f v = *(const v2f*)&rowp[e];
            mx = fmaxf(mx, fmaxf(v.x, v.y));
        }
        float tv[TOPK];
        int   ti[TOPK];
#pragma unroll
        for (int j = 0; j < TOPK; ++j) { tv[j] = -3.4e38f; ti[j] = 0; }
        float sum = 0.f;
        for (int e = 0; e < NEXP; ++e) {
            const float v = rowp[e];
            sum += __expf(v - mx);
            if (v > tv[TOPK - 1]) {
#pragma unroll
                for (int j = 0; j < TOPK; ++j) {
                    if (v > tv[j]) {   // strict > keeps lower index first on ties (jax top_k)
#pragma unroll
                        for (int q = TOPK - 1; q > j; --q) {
                            tv[q] = tv[q - 1];
                            ti[q] = ti[q - 1];
                        }
                        tv[j] = v;
                        ti[j] = e;
                        break;
                    }
                }
            }
        }
        const float inv = 1.f / sum;
        const int tok = tok0 + tid;
#pragma unroll
        for (int j = 0; j < TOPK; ++j) {
            out_w[tok * TOPK + j] = __expf(tv[j] - mx) * inv;
            out_i[tok * TOPK + j] = ti[j];
        }
    }
}

extern "C" void kernel_launch(void* const* d_in, const int* in_sizes, int n_in,
                              void* d_out, int out_size, void* d_ws, size_t ws_size,
                              hipStream_t stream) {
    const float* x = (const float*)d_in[0];   // [32768, 4096] fp32
    const float* w = (const float*)d_in[1];   // [128, 4096] fp32
    float* out_w = (float*)d_out;                                  // [32768, 8] fp32
    int*   out_i = (int*)((float*)d_out + (size_t)TOKENS * TOPK);  // [32768, 8] int32 bits
    dim3 grid(TOKENS / BM);
    dim3 block(256);
    router_kernel<<<grid, block, 0, stream>>>(x, w, out_w, out_i);
}